// SecondaryCapsuleLayer_5789615915069
// MI455X (gfx1250) — compile-verified
//
#include <hip/hip_runtime.h>

typedef float v2f __attribute__((ext_vector_type(2)));
typedef float v8f __attribute__((ext_vector_type(8)));

#define B_DIM 64
#define I_DIM 8
#define C_DIM 2048
#define U_DIM 32
#define D_DIM 16
#define K_DIM (C_DIM * I_DIM)   // 16384, k = i*C_DIM + c
#define N_DIM (U_DIM * D_DIM)   // 512
#define W_SU  (U_DIM * D_DIM * I_DIM)  // 4096 (stride over c)
#define W_SD  (D_DIM * I_DIM)          // 128  (stride over u)

// ---------------- init b_ij = 0 ----------------
__global__ void init_b_kernel(float* b_ij) {
  int t = blockIdx.x * blockDim.x + threadIdx.x;
  if (t < C_DIM * U_DIM) b_ij[t] = 0.0f;
}

// ---------------- softmax over U per c-row (one wave32 per row) ----------------
__global__ __launch_bounds__(32) void softmax_kernel(const float* __restrict__ b_ij,
                                                     float* __restrict__ c_ij) {
  int c = blockIdx.x;
  int u = threadIdx.x;  // 0..31
  float val = b_ij[c * U_DIM + u];
  float m = val;
  for (int off = 16; off > 0; off >>= 1)
    m = fmaxf(m, __shfl_xor(m, off, 32));
  float e = __expf(val - m);
  float ssum = e;
  for (int off = 16; off > 0; off >>= 1)
    ssum += __shfl_xor(ssum, off, 32);
  c_ij[c * U_DIM + u] = e / ssum;
}

// ---------------- s[b,u,d] = sum_{c,i} x[b,i,c] * c_ij[c,u] * W[c,u,d,i] ----------
// GEMM M=64(b) N=512(u,d) K=16384(i,c). Grid: (4 b-tiles, 32 u). Block: 8 waves,
// K split across waves, v_wmma_f32_16x16x4_f32 accumulate, LDS cross-wave reduce.
__global__ __launch_bounds__(256) void s_gemm_kernel(const float* __restrict__ x,
                                                     const float* __restrict__ W,
                                                     const float* __restrict__ c_ij,
                                                     float* __restrict__ s) {
  const int u    = blockIdx.y;
  const int b0   = blockIdx.x * 16;
  const int lane = threadIdx.x & 31;
  const int wave = threadIdx.x >> 5;
  const int half = lane >> 4;   // 0/1
  const int lm   = lane & 15;   // A: row m (b); B: col n (d)

  v8f acc = {};
  const float* xrow = x + (size_t)(b0 + lm) * K_DIM;  // x[b] flat over (i,c) == k
  const int kbase = wave * (K_DIM / 8);               // 2048 per wave

  for (int step = 0; step < (K_DIM / 8) / 4; ++step) {
    const int koff = kbase + step * 4 + half * 2;  // even; i constant over koff..koff+1
    v2f a, bf;
    a.x = xrow[koff];
    a.y = xrow[koff + 1];
    const int c0 = koff & (C_DIM - 1);
    const int i0 = koff >> 11;
    const float* wp = W + (size_t)c0 * W_SU + u * W_SD + lm * I_DIM + i0;
    bf.x = wp[0]    * c_ij[c0 * U_DIM + u];
    bf.y = wp[W_SU] * c_ij[(c0 + 1) * U_DIM + u];
    __builtin_prefetch(xrow + koff + 64, 0, 1);
    acc = __builtin_amdgcn_wmma_f32_16x16x4_f32(false, a, false, bf,
                                                (short)0, acc, false, false);
  }

  // cross-wave reduction in LDS
  __shared__ float red[8][32][8];
  for (int r = 0; r < 8; ++r) red[wave][lane][r] = acc[r];
  __syncthreads();

  const int t = threadIdx.x;      // 256 threads: one (lane,r) pair each
  const int l = t & 31;
  const int r = t >> 5;           // 0..7
  float sum = 0.0f;
  for (int w = 0; w < 8; ++w) sum += red[w][l][r];
  const int m = r + 8 * (l >> 4);
  const int n = l & 15;
  s[(size_t)(b0 + m) * N_DIM + u * D_DIM + n] = sum;
}

// ---------------- squash (faithful quirk: magnitude summed over U axis) ----------
__global__ __launch_bounds__(512) void squash_kernel(const float* __restrict__ s,
                                                     float* __restrict__ v) {
  const int b = blockIdx.x;
  const int t = threadIdx.x;  // t = u*16 + d
  const int d = t & 15;
  const float sv = s[(size_t)b * N_DIM + t];
  __shared__ float sq[512];
  sq[t] = sv * sv;
  __syncthreads();
  for (int stride = 256; stride >= 16; stride >>= 1) {
    if (t < stride) sq[t] += sq[t + stride];
    __syncthreads();
  }
  const float mag_sq = sq[d];   // sum over u for this (b,d)
  const float mag = sqrtf(mag_sq);
  v[(size_t)b * N_DIM + t] = mag_sq / (1.0f + mag_sq) * (sv / mag);
}

// ---- b_ij[c,u] += (1/B) * sum_{i,d} W[c,u,d,i] * (sum_b x[b,i,c] * v[b,u,d]) ----
// Per block: one wave, 16-c tile x one u. The v tile (64 b x 16 d, 4KB) is staged
// once into LDS with GLOBAL_LOAD_ASYNC_TO_LDS_B128 (ASYNCcnt) instead of being
// re-fetched from global on every i iteration. Then: M-tile (16 c x 16 d) = X^T V
// via 16 WMMA steps over batch (K=64), contracted with W into LDS via ds_add_f32.
__global__ __launch_bounds__(32) void delta_kernel(const float* __restrict__ x,
                                                   const float* __restrict__ W,
                                                   const float* __restrict__ v,
                                                   float* __restrict__ b_ij) {
  const int u    = blockIdx.y;
  const int c0   = blockIdx.x * 16;
  const int lane = threadIdx.x;
  const int half = lane >> 4;
  const int lm   = lane & 15;   // A: row (c local); B: col (d)

  __shared__ float vtile[B_DIM * D_DIM];  // [b][d] for this block's u (4KB)
  __shared__ float cacc[16];

  // --- async stage v[:, u, :] -> LDS: 256 chunks of 16B, 8 per lane ---
  // Generic shared pointer's low 32 bits are the wave-relative LDS byte address
  const unsigned int vbase = (unsigned int)(uintptr_t)vtile;
  for (int it2 = 0; it2 < 8; ++it2) {
    const int chunk = it2 * 32 + lane;            // 0..255
    const int bb = chunk >> 2;                    // b index
    const int seg = chunk & 3;                    // 4-float segment of d
    const float* gptr = v + (size_t)bb * N_DIM + u * D_DIM + seg * 4;
    const unsigned int lds_off = vbase + (unsigned int)chunk * 16u;
    asm volatile("global_load_async_to_lds_b128 %0, %1, off"
                 :: "v"(lds_off), "v"(gptr)
                 : "memory");
  }

  if (lane < 16) cacc[lane] = 0.0f;
  asm volatile("s_wait_asynccnt 0x0" ::: "memory");
  __syncthreads();

  for (int i = 0; i < I_DIM; ++i) {
    v8f acc = {};
    for (int bs = 0; bs < B_DIM / 4; ++bs) {
      const int bb = bs * 4 + half * 2;
      v2f a, bf;
      a.x  = x[(size_t)bb       * K_DIM + i * C_DIM + c0 + lm];
      a.y  = x[(size_t)(bb + 1) * K_DIM + i * C_DIM + c0 + lm];
      bf.x = vtile[bb * D_DIM + lm];
      bf.y = vtile[(bb + 1) * D_DIM + lm];
      acc = __builtin_amdgcn_wmma_f32_16x16x4_f32(false, a, false, bf,
                                                  (short)0, acc, false, false);
    }
    // acc[r] = M[c = c0 + r + 8*half, d = lm] for this i; contract with W over d
    for (int r = 0; r < 8; ++r) {
      const int cl = r + 8 * half;
      const float wv = W[(size_t)(c0 + cl) * W_SU + u * W_SD + lm * I_DIM + i];
      atomicAdd(&cacc[cl], acc[r] * wv);
    }
  }
  __syncthreads();
  if (lane < 16)
    b_ij[(c0 + lane) * U_DIM + u] += cacc[lane] * (1.0f / B_DIM);
}

extern "C" void kernel_launch(void* const* d_in, const int* in_sizes, int n_in,
                              void* d_out, int out_size, void* d_ws, size_t ws_size,
                              hipStream_t stream) {
  const float* x = (const float*)d_in[0];   // (B, I, C)
  const float* W = (const float*)d_in[1];   // (C, U, D, I)
  float* out = (float*)d_out;               // (B, U, D, 1) contiguous

  float* b_ij = (float*)d_ws;               // (C, U)
  float* c_ij = b_ij + C_DIM * U_DIM;       // (C, U)
  float* s    = c_ij + C_DIM * U_DIM;       // (B, U, D)
  float* v    = s + B_DIM * N_DIM;          // (B, U, D)

  init_b_kernel<<<(C_DIM * U_DIM + 255) / 256, 256, 0, stream>>>(b_ij);

  for (int it = 0; it < 3; ++it) {
    softmax_kernel<<<C_DIM, 32, 0, stream>>>(b_ij, c_ij);

    dim3 gs(4, U_DIM);
    s_gemm_kernel<<<gs, 256, 0, stream>>>(x, W, c_ij, s);

    float* vdst = (it == 2) ? out : v;
    squash_kernel<<<B_DIM, 512, 0, stream>>>(s, vdst);

    if (it < 2) {
      dim3 gd(C_DIM / 16, U_DIM);
      delta_kernel<<<gd, 32, 0, stream>>>(x, W, v, b_ij);
    }
  }
}